// AtomCenteredTensorMomentDescriptor_88502096101736
// MI455X (gfx1250) — compile-verified
//
#include <hip/hip_runtime.h>
#include <math.h>

// ---------------------------------------------------------------------------
// AtomCenteredTensorMomentDescriptor on gfx1250 (MI455X), fp32 WMMA pipeline.
//
// Weighted-CG tables G1/G2 are built on device into d_ws (~1.75 MB), stored
// K-pair-interleaved so WMMA B-operands load as a single global_load_b64.
// Both tensor-product layers run as per-radial-channel GEMMs on
// V_WMMA_F32_16X16X4_F32. Each wave processes TWO 16-edge tiles so every
// B-operand load feeds two WMMAs (halved vmem per matrix op), and the basis
// stage uses all 32 lanes (lanes 0-15 -> tile0, 16-31 -> tile1).
// ---------------------------------------------------------------------------

#define E_EDGES   131072
#define NUM_F     16
#define NP1       40      // paths layer 1 (lin=3, lout=4)
#define NP2       65      // paths layer 2 (lin=4, lout=4)
#define K1DIM     256     // 16*16 pair indices, layer 1
#define K2DIM     640     // 25*25 = 625 padded to multiple of 16
#define NPAD      32      // output dim 25 padded to 2 WMMA N-blocks
#define G1_FLOATS ((size_t)NUM_F * K1DIM * NPAD)   // 131072
#define G2_FLOATS ((size_t)NUM_F * K2DIM * NPAD)   // 327680

typedef float v2f __attribute__((ext_vector_type(2)));
typedef float v8f __attribute__((ext_vector_type(8)));

// ============================= CG construction =============================

__device__ __forceinline__ double dfact(int n) {
  const double t[14] = {1.0, 1.0, 2.0, 6.0, 24.0, 120.0, 720.0, 5040.0,
                        40320.0, 362880.0, 3628800.0, 39916800.0,
                        479001600.0, 6227020800.0};
  return t[n];
}

__device__ double cg_coeff(int j1, int m1, int j2, int m2, int j3, int m3) {
  if (m1 + m2 != m3) return 0.0;
  if (j3 < abs(j1 - j2) || j3 > j1 + j2) return 0.0;
  double pre = sqrt((double)(2 * j3 + 1) * dfact(j3 + j1 - j2) *
                    dfact(j3 - j1 + j2) * dfact(j1 + j2 - j3) /
                    dfact(j1 + j2 + j3 + 1));
  pre *= sqrt(dfact(j3 + m3) * dfact(j3 - m3) * dfact(j1 - m1) *
              dfact(j1 + m1) * dfact(j2 - m2) * dfact(j2 + m2));
  int klo = max(0, max(j2 - j3 - m1, j1 - j3 + m2));
  int khi = min(j1 + j2 - j3, min(j1 - m1, j2 + m2));
  double s = 0.0;
  for (int k = klo; k <= khi; ++k) {
    double d = dfact(k) * dfact(j1 + j2 - j3 - k) * dfact(j1 - m1 - k) *
               dfact(j2 + m2 - k) * dfact(j3 - j2 + m1 + k) *
               dfact(j3 - j1 - m2 + k);
    s += ((k & 1) ? -1.0 : 1.0) / d;
  }
  return pre * s;
}

// Rows of the complex->real unitary U(l): at most 2 nonzero (col, value).
__device__ __forceinline__ int uterms(int l, int i, int* col, double* vr,
                                      double* vi) {
  const double s2 = 0.70710678118654752440;
  int mm = i - l;
  if (mm == 0) { col[0] = l; vr[0] = 1.0; vi[0] = 0.0; return 1; }
  if (mm > 0) {
    double sgn = (mm & 1) ? -1.0 : 1.0;
    col[0] = l + mm; vr[0] = sgn * s2; vi[0] = 0.0;
    col[1] = l - mm; vr[1] = s2;       vi[1] = 0.0;
    return 2;
  }
  int m = -mm;
  double sgn = (m & 1) ? -1.0 : 1.0;
  col[0] = l - m; vr[0] = 0.0; vi[0] = s2;
  col[1] = l + m; vr[1] = 0.0; vi[1] = -sgn * s2;
  return 2;
}

// M[i,j,k] = sum_{a,b,c} U1[i,a] U2[j,b] conj(U3[k,c]) CG(a,b,c)
__device__ void real_cg_entry(int l1, int l2, int l3, int i, int j, int k,
                              double& Re, double& Im) {
  int ca[2], cb[2], cc[2];
  double ar[2], ai[2], br[2], bi[2], cr[2], ci[2];
  int na = uterms(l1, i, ca, ar, ai);
  int nb = uterms(l2, j, cb, br, bi);
  int nc = uterms(l3, k, cc, cr, ci);
  Re = 0.0; Im = 0.0;
  for (int a = 0; a < na; ++a)
    for (int b = 0; b < nb; ++b)
      for (int c = 0; c < nc; ++c) {
        double g = cg_coeff(l1, ca[a] - l1, l2, cb[b] - l2, l3, cc[c] - l3);
        if (g == 0.0) continue;
        double tr = ar[a] * br[b] - ai[a] * bi[b];
        double ti = ar[a] * bi[b] + ai[a] * br[b];
        double rr = tr * cr[c] + ti * ci[c];   // * conj(u3)
        double ii = ti * cr[c] - tr * ci[c];
        Re += g * rr; Im += g * ii;
      }
}

__device__ __forceinline__ void path_lookup(int lin, int lout, int idx,
                                            int& L1, int& L2, int& L3) {
  int c = 0;
  for (int l1 = 0; l1 <= lin; ++l1)
    for (int l2 = 0; l2 <= lin; ++l2) {
      int lo = abs(l1 - l2), hi = min(l1 + l2, lout);
      for (int l3 = lo; l3 <= hi; ++l3) {
        if (c == idx) { L1 = l1; L2 = l2; L3 = l3; return; }
        ++c;
      }
    }
  L1 = L2 = L3 = 0;
}

__global__ void zero_g_kernel(float* g, int n) {
  int i = blockIdx.x * blockDim.x + threadIdx.x;
  if (i < n) g[i] = 0.0f;
}

// One block per path. Builds the K-pair-interleaved weighted-CG tables:
//   G[f-slice][(ij>>1)*(NPAD*2) + kf*2 + (ij&1)] = w[p,f] * realCG[i,j,k]
// Paths occupy disjoint (l1,l2,l3) blocks -> no write races.
__global__ void build_g_kernel(const float* __restrict__ w1,
                               const float* __restrict__ w2,
                               float* __restrict__ G1,
                               float* __restrict__ G2) {
  int b = blockIdx.x;
  int tid = threadIdx.x;
  int layer2 = (b >= NP1);
  int pidx = layer2 ? (b - NP1) : b;
  int L1, L2, L3;
  if (layer2) path_lookup(4, 4, pidx, L1, L2, L3);
  else        path_lookup(3, 4, pidx, L1, L2, L3);
  int n1 = 2 * L1 + 1, n2 = 2 * L2 + 1, n3 = 2 * L3 + 1;
  int n = n1 * n2 * n3;

  __shared__ double sR[64], sI[64];
  __shared__ int sPick;
  double aR = 0.0, aI = 0.0;
  for (int idx = tid; idx < n; idx += 64) {
    int k = idx % n3, t = idx / n3, j = t % n2, i = t / n2;
    double Re, Im;
    real_cg_entry(L1, L2, L3, i, j, k, Re, Im);
    aR += fabs(Re); aI += fabs(Im);
  }
  sR[tid] = aR; sI[tid] = aI;
  __syncthreads();
  if (tid == 0) {
    double tR = 0.0, tI = 0.0;
    for (int t = 0; t < 64; ++t) { tR += sR[t]; tI += sI[t]; }
    sPick = (tR >= tI) ? 1 : 0;
  }
  __syncthreads();
  int useR = sPick;

  const float* w = layer2 ? w2 : w1;
  float* G = layer2 ? G2 : G1;
  int Din = layer2 ? 25 : 16;
  int Kd  = layer2 ? K2DIM : K1DIM;
  for (int idx = tid; idx < n; idx += 64) {
    int k = idx % n3, t = idx / n3, j = t % n2, i = t / n2;
    double Re, Im;
    real_cg_entry(L1, L2, L3, i, j, k, Re, Im);
    float v = (float)(useR ? Re : Im);
    int ij = (L1 * L1 + i) * Din + (L2 * L2 + j);
    int kf = L3 * L3 + k;
    for (int f = 0; f < NUM_F; ++f) {
      G[(size_t)f * Kd * NPAD +
        (size_t)(ij >> 1) * (NPAD * 2) + kf * 2 + (ij & 1)] =
          w[pidx * NUM_F + f] * v;
    }
  }
}

// ============================== main kernel ================================
// grid = (64, 16): blockIdx.y = radial channel f. 256 threads = 8 waves;
// each wave owns a PAIR of 16-edge tiles per iteration (B reused 2x).

__launch_bounds__(256)
__global__ void moment_main_kernel(const float* __restrict__ disp,
                                   const float* __restrict__ G1,
                                   const float* __restrict__ G2,
                                   float* __restrict__ out) {
  constexpr float PI = 3.14159265358979323846f;
  __shared__ float sY[8 * 2 * 16 * 16];   // per-wave basis tiles  [t][i][e]
  __shared__ float sT[8 * 2 * 32 * 16];   // per-wave layer1 tiles [t][k][e]

  const int tid  = threadIdx.x;
  const int wave = tid >> 5;
  const int lane = tid & 31;
  const int half = lane >> 4;     // WMMA K/M lane-half select
  const int m    = lane & 15;
  const int f    = blockIdx.y;
  const int y0 = wave * 512;            // tile0 basis
  const int y1 = wave * 512 + 256;      // tile1 basis
  const int t0 = wave * 1024;           // tile0 T
  const int t1 = wave * 1024 + 512;     // tile1 T

  // K-pair-interleaved tables, viewed as v2f: element (p, n) at p*NPAD + n
  // holds { G[ij=2p][n], G[ij=2p+1][n] }  -> one global_load_b64 per B operand.
  const v2f* G1p = (const v2f*)(G1 + (size_t)f * K1DIM * NPAD);
  const v2f* G2p = (const v2f*)(G2 + (size_t)f * K2DIM * NPAD);

  const int pairsTotal = E_EDGES / 32;          // 4096
  const int stride = gridDim.x * 8;

  for (int pair = blockIdx.x * 8 + wave; pair < pairsTotal; pair += stride) {
    // prefetch next pair's displacements (global_prefetch_b8)
    {
      int np = pair + stride; if (np >= pairsTotal) np = pair;
      __builtin_prefetch(disp + (size_t)np * 96, 0, 0);
    }

    // ---- basis: all 32 lanes; lane half selects which tile's edge ----
    {
      int e = pair * 32 + half * 16 + m;
      float dx = disp[3 * e + 0], dy = disp[3 * e + 1], dz = disp[3 * e + 2];
      float r = sqrtf(dx * dx + dy * dy + dz * dz);
      float inv = 1.0f / fmaxf(r, 1e-9f);
      float ux = dx * inv, uy = dy * inv, uz = dz * inv;
      float xx = ux * ux, yy = uy * uy, zz = uz * uz;
      float xr = r * 0.2f;                       // r / CUTOFF
      float t = PI * xr * (float)(f + 1);
      float rb = (fabsf(t) < 1e-8f) ? 1.0f : (sinf(t) / t);   // jnp.sinc
      float cut = 0.0f;
      if (xr < 1.0f) { float x2 = xr * xr; cut = expf(1.0f - 1.0f / (1.0f - x2)); }
      float s = cut * rb;
      float sh[16];
      sh[0]  = 0.28209479177387814f;
      sh[1]  = 0.4886025119029199f * uy;
      sh[2]  = 0.4886025119029199f * uz;
      sh[3]  = 0.4886025119029199f * ux;
      sh[4]  = 1.0925484305920792f * ux * uy;
      sh[5]  = 1.0925484305920792f * uy * uz;
      sh[6]  = 0.31539156525252005f * (3.0f * zz - 1.0f);
      sh[7]  = 1.0925484305920792f * ux * uz;
      sh[8]  = 0.5462742152960396f * (xx - yy);
      sh[9]  = 0.5900435899266435f * uy * (3.0f * xx - yy);
      sh[10] = 2.890611442640554f * ux * uy * uz;
      sh[11] = 0.4570457994644658f * uy * (5.0f * zz - 1.0f);
      sh[12] = 0.3731763325901154f * uz * (5.0f * zz - 3.0f);
      sh[13] = 0.4570457994644658f * ux * (5.0f * zz - 1.0f);
      sh[14] = 1.445305721320277f * uz * (xx - yy);
      sh[15] = 0.5900435899266435f * ux * (xx - 3.0f * yy);
#pragma unroll
      for (int i = 0; i < 16; ++i)
        sY[y0 + half * 256 + i * 16 + m] = s * sh[i];
    }
    asm volatile("s_wait_dscnt 0" ::: "memory");  // wave-local LDS handoff

    // ---- layer 1: Z[e,k] = sum_ij (y_i*y_j) * G1[f][ij][k], K=256 ----
    v8f aA0 = {}, aA1 = {}, aB0 = {}, aB1 = {};
#pragma unroll 4
    for (int c = 0; c < K1DIM / 4; ++c) {
      int ij0 = c * 4 + 2 * half;               // lane's K element (even)
      int p   = c * 2 + half;                   // pair index = ij0 >> 1
      int i0 = ij0 >> 4, j0 = ij0 & 15;
      int i1 = (ij0 + 1) >> 4, j1 = (ij0 + 1) & 15;
      v2f A0, A1;
      A0.x = sY[y0 + i0 * 16 + m] * sY[y0 + j0 * 16 + m];
      A0.y = sY[y0 + i1 * 16 + m] * sY[y0 + j1 * 16 + m];
      A1.x = sY[y1 + i0 * 16 + m] * sY[y1 + j0 * 16 + m];
      A1.y = sY[y1 + i1 * 16 + m] * sY[y1 + j1 * 16 + m];
      v2f B0 = G1p[p * NPAD + m];
      v2f B1 = G1p[p * NPAD + 16 + m];
      aA0 = __builtin_amdgcn_wmma_f32_16x16x4_f32(false, A0, false, B0,
                                                  (short)0, aA0, false, false);
      aA1 = __builtin_amdgcn_wmma_f32_16x16x4_f32(false, A0, false, B1,
                                                  (short)0, aA1, false, false);
      aB0 = __builtin_amdgcn_wmma_f32_16x16x4_f32(false, A1, false, B0,
                                                  (short)0, aB0, false, false);
      aB1 = __builtin_amdgcn_wmma_f32_16x16x4_f32(false, A1, false, B1,
                                                  (short)0, aB1, false, false);
    }
    // scatter D (rows = edges, cols = k) into T[k][e]; k>=25 columns are
    // exact zeros (G1 pad), so T rows 25..31 are clean zero padding.
#pragma unroll
    for (int r = 0; r < 8; ++r) {
      sT[t0 + (m)      * 16 + (r + 8 * half)] = aA0[r];
      sT[t0 + (16 + m) * 16 + (r + 8 * half)] = aA1[r];
      sT[t1 + (m)      * 16 + (r + 8 * half)] = aB0[r];
      sT[t1 + (16 + m) * 16 + (r + 8 * half)] = aB1[r];
    }
    asm volatile("s_wait_dscnt 0" ::: "memory");

    // ---- layer 2: out[e,k] = sum_ij (T_i*T_j) * G2[f][ij][k], K=640 ----
    v8f oA0 = {}, oA1 = {}, oB0 = {}, oB1 = {};
#pragma unroll 2
    for (int c = 0; c < K2DIM / 4; ++c) {
      int ij0 = c * 4 + 2 * half;
      int p   = c * 2 + half;
      int i0 = ij0 / 25, j0 = ij0 - i0 * 25;
      int i1 = (ij0 + 1) / 25, j1 = (ij0 + 1) - i1 * 25;
      v2f A0, A1;
      A0.x = sT[t0 + i0 * 16 + m] * sT[t0 + j0 * 16 + m];
      A0.y = sT[t0 + i1 * 16 + m] * sT[t0 + j1 * 16 + m];
      A1.x = sT[t1 + i0 * 16 + m] * sT[t1 + j0 * 16 + m];
      A1.y = sT[t1 + i1 * 16 + m] * sT[t1 + j1 * 16 + m];
      v2f B0 = G2p[p * NPAD + m];
      v2f B1 = G2p[p * NPAD + 16 + m];
      oA0 = __builtin_amdgcn_wmma_f32_16x16x4_f32(false, A0, false, B0,
                                                  (short)0, oA0, false, false);
      oA1 = __builtin_amdgcn_wmma_f32_16x16x4_f32(false, A0, false, B1,
                                                  (short)0, oA1, false, false);
      oB0 = __builtin_amdgcn_wmma_f32_16x16x4_f32(false, A1, false, B0,
                                                  (short)0, oB0, false, false);
      oB1 = __builtin_amdgcn_wmma_f32_16x16x4_f32(false, A1, false, B1,
                                                  (short)0, oB1, false, false);
    }

    // ---- store: out[e, k, f], shape (E, 25, 16) ----
    int ebase = pair * 32;
#pragma unroll
    for (int r = 0; r < 8; ++r) {
      int e0 = ebase + r + 8 * half;        // tile0 edges
      int e1 = ebase + 16 + r + 8 * half;   // tile1 edges
      out[((size_t)e0 * 25 + m) * 16 + f] = oA0[r];
      out[((size_t)e1 * 25 + m) * 16 + f] = oB0[r];
      if (m < 9) {
        out[((size_t)e0 * 25 + 16 + m) * 16 + f] = oA1[r];
        out[((size_t)e1 * 25 + 16 + m) * 16 + f] = oB1[r];
      }
    }
  }
}

// ================================ launcher =================================

extern "C" void kernel_launch(void* const* d_in, const int* in_sizes, int n_in,
                              void* d_out, int out_size, void* d_ws,
                              size_t ws_size, hipStream_t stream) {
  // inputs: [0]=atomic_numbers (unused), [1]=neighbour_displacements (E*3 f32),
  //         [2]=neighbour_indices (unused), [3]=tp_weights1 (40*16 f32),
  //         [4]=tp_weights2 (65*16 f32)
  const float* disp = (const float*)d_in[1];
  const float* w1   = (const float*)d_in[3];
  const float* w2   = (const float*)d_in[4];
  float* out = (float*)d_out;

  float* G1 = (float*)d_ws;                  // needs ~1.75 MB of workspace
  float* G2 = G1 + G1_FLOATS;
  int totalG = (int)(G1_FLOATS + G2_FLOATS); // 458752 floats

  zero_g_kernel<<<dim3((totalG + 255) / 256), dim3(256), 0, stream>>>(G1, totalG);
  build_g_kernel<<<dim3(NP1 + NP2), dim3(64), 0, stream>>>(w1, w2, G1, G2);
  moment_main_kernel<<<dim3(64, NUM_F), dim3(256), 0, stream>>>(disp, G1, G2, out);
}